// GCN_83356725280990
// MI455X (gfx1250) — compile-verified
//
#include <hip/hip_runtime.h>
#include <hip/hip_bf16.h>

// ---------------------------------------------------------------------------
// GCN forward on MI455X (gfx1250). fp32 end-to-end; GEMMs use
// V_WMMA_F32_16X16X4_F32 (native fp32 WMMA, wave32).
// Each wave computes a 16x64 output strip (4 WMMA accumulators) so the
// A-operand float2 load is amortized over 4 WMMAs.
// ---------------------------------------------------------------------------

typedef float v2f __attribute__((ext_vector_type(2)));
typedef float v8f __attribute__((ext_vector_type(8)));

#define N_NODES 10000
#define N_EDGES 320000
#define IN_F    128
#define HID_F   256

// ---------------------------------------------------------------------------
// C[M x F] = A[M x K] @ B[K x F], row-major fp32.
// A-matrix VGPR layout (ISA 7.12.2, 32-bit A 16x4):
//   lanes 0-15 : v0 = A[m][k+0], v1 = A[m][k+1]
//   lanes 16-31: v0 = A[m][k+2], v1 = A[m][k+3]
// B-matrix (4x16, row striped across lanes, half-split like C/D):
//   v0: lanes 0-15 = B[k+0][n], lanes 16-31 = B[k+2][n]
//   v1: lanes 0-15 = B[k+1][n], lanes 16-31 = B[k+3][n]
// C/D: VGPR r -> row (r + 8*hi), col = lane%16.
// ---------------------------------------------------------------------------
__global__ __launch_bounds__(256) void gemm_wmma_f32(
    const float* __restrict__ A, const float* __restrict__ B,
    float* __restrict__ C, int M, int K, int F)
{
    const int lane    = threadIdx.x & 31;
    const int wave    = threadIdx.x >> 5;
    const int stripsN = F >> 6;                      // 64-wide N strips
    const int task    = blockIdx.x * 8 + wave;
    if (task >= (M >> 4) * stripsN) return;
    const int tm = task / stripsN;
    const int ts = task % stripsN;
    const int n0 = ts * 64;

    const int hi = lane >> 4;      // 0: lanes 0-15, 1: lanes 16-31
    const int lr = lane & 15;

    const float* __restrict__ Arow  = A + (size_t)(tm * 16 + lr) * K + 2 * hi;
    const float* __restrict__ Bbase = B + (size_t)(2 * hi) * F + n0 + lr;

    v8f acc0 = {}, acc1 = {}, acc2 = {}, acc3 = {};
    for (int k = 0; k < K; k += 4) {
        v2f a;
        a.x = Arow[k];                      // A[m][k + 2*hi]
        a.y = Arow[k + 1];                  // A[m][k + 2*hi + 1]
        const float* __restrict__ Bk = Bbase + (size_t)k * F;
        v2f b0, b1, b2, b3;
        b0.x = Bk[0];       b0.y = Bk[(size_t)F + 0];
        b1.x = Bk[16];      b1.y = Bk[(size_t)F + 16];
        b2.x = Bk[32];      b2.y = Bk[(size_t)F + 32];
        b3.x = Bk[48];      b3.y = Bk[(size_t)F + 48];
        acc0 = __builtin_amdgcn_wmma_f32_16x16x4_f32(false, a, false, b0,
                                                     (short)0, acc0, false, false);
        acc1 = __builtin_amdgcn_wmma_f32_16x16x4_f32(false, a, false, b1,
                                                     (short)0, acc1, false, false);
        acc2 = __builtin_amdgcn_wmma_f32_16x16x4_f32(false, a, false, b2,
                                                     (short)0, acc2, false, false);
        acc3 = __builtin_amdgcn_wmma_f32_16x16x4_f32(false, a, false, b3,
                                                     (short)0, acc3, false, false);
    }

    float* __restrict__ Crow = C + (size_t)(tm * 16 + 8 * hi) * F + n0 + lr;
#pragma unroll
    for (int r = 0; r < 8; ++r) {
        Crow[(size_t)r * F +  0] = acc0[r];
        Crow[(size_t)r * F + 16] = acc1[r];
        Crow[(size_t)r * F + 32] = acc2[r];
        Crow[(size_t)r * F + 48] = acc3[r];
    }
}

// out-degree of src (float counts; exact up to 2^24)
__global__ void degree_kernel(const int* __restrict__ src,
                              float* __restrict__ deg, int E)
{
    int i = blockIdx.x * blockDim.x + threadIdx.x;
    if (i < E) atomicAdd(&deg[src[i]], 1.0f);
}

__global__ void invdeg_kernel(const float* __restrict__ deg,
                              float* __restrict__ inv, int N)
{
    int i = blockIdx.x * blockDim.x + threadIdx.x;
    if (i < N) {
        float d = deg[i];
        inv[i] = (d > 0.0f) ? (1.0f / d) : 0.0f;
    }
}

// agg[dst[e]][:] += hW[src[e]][:] * inv_deg[src[e]]
// 4 edges per 256-thread block; 64 threads (float4 each) cover F == 256.
__global__ __launch_bounds__(256) void edge_scatter(
    const float* __restrict__ hW, const int* __restrict__ src,
    const int* __restrict__ dst, const float* __restrict__ inv_deg,
    float* __restrict__ agg, int E, int F)
{
    const int e = blockIdx.x * 4 + (threadIdx.x >> 6);
    if (e >= E) return;
    const int t = threadIdx.x & 63;

    const int s = src[e];
    const int d = dst[e];
    const float w = inv_deg[s];

    const float4* __restrict__ ps = (const float4*)(hW + (size_t)s * F);
    float4 v = ps[t];
    float* __restrict__ pd = agg + (size_t)d * F + t * 4;
    atomicAdd(pd + 0, v.x * w);
    atomicAdd(pd + 1, v.y * w);
    atomicAdd(pd + 2, v.z * w);
    atomicAdd(pd + 3, v.w * w);
}

__global__ void bias_act_kernel(float* __restrict__ h,
                                const float* __restrict__ b,
                                int total, int F, int relu)
{
    int i = blockIdx.x * blockDim.x + threadIdx.x;
    if (i < total) {
        float v = h[i] + b[i % F];
        h[i] = relu ? fmaxf(v, 0.0f) : v;
    }
}

// partial column sums: block p sums rows p, p+P, ... into part[p*F + f]
__global__ __launch_bounds__(256) void colsum_partial(
    const float* __restrict__ h, float* __restrict__ part, int N, int F)
{
    const int f = threadIdx.x;  // F == 256 == blockDim.x
    const int p = blockIdx.x;
    float s = 0.0f;
    for (int n = p; n < N; n += gridDim.x)
        s += h[(size_t)n * F + f];
    part[(size_t)p * F + f] = s;
}

// V = mean_nodes(h) @ Wv + bv   (single block, 256 threads)
__global__ __launch_bounds__(256) void value_head(
    const float* __restrict__ part, int P,
    const float* __restrict__ Wv, const float* __restrict__ bv,
    float* __restrict__ outV, int N, int F)
{
    const int f = threadIdx.x;
    float s = 0.0f;
    for (int p = 0; p < P; ++p)
        s += part[(size_t)p * F + f];
    float m = s / (float)N;
    __shared__ float red[256];
    red[f] = m * Wv[f];
    __syncthreads();
    for (int off = 128; off > 0; off >>= 1) {
        if (f < off) red[f] += red[f + off];
        __syncthreads();
    }
    if (f == 0) outV[0] = red[0] + bv[0];
}

// PI[n] = h[n] . Wp + bp ; one wave32 per node, 8 waves per block
__global__ __launch_bounds__(256) void policy_head(
    const float* __restrict__ h, const float* __restrict__ Wp,
    const float* __restrict__ bp, float* __restrict__ PI, int N, int F)
{
    const int wave = threadIdx.x >> 5;
    const int lane = threadIdx.x & 31;
    const int n = blockIdx.x * 8 + wave;
    if (n >= N) return;
    const float* __restrict__ row = h + (size_t)n * F;
    float s = 0.0f;
    for (int f = lane; f < F; f += 32)
        s += row[f] * Wp[f];
#pragma unroll
    for (int off = 16; off > 0; off >>= 1)
        s += __shfl_down(s, off, 32);
    if (lane == 0) PI[n] = s + bp[0];
}

// ---------------------------------------------------------------------------
// Host-side orchestration (all stream-ordered; graph-capture safe).
// ---------------------------------------------------------------------------
extern "C" void kernel_launch(void* const* d_in, const int* in_sizes, int n_in,
                              void* d_out, int out_size, void* d_ws, size_t ws_size,
                              hipStream_t stream)
{
    const float* features = (const float*)d_in[0];   // [N, 128]
    const int*   src      = (const int*)  d_in[1];   // [E]
    const int*   dst      = (const int*)  d_in[2];   // [E]
    const float* W1 = (const float*)d_in[3];  const float* b1 = (const float*)d_in[4];
    const float* W2 = (const float*)d_in[5];  const float* b2 = (const float*)d_in[6];
    const float* W3 = (const float*)d_in[7];  const float* b3 = (const float*)d_in[8];
    const float* Wp = (const float*)d_in[9];  const float* bp = (const float*)d_in[10];
    const float* Wv = (const float*)d_in[11]; const float* bv = (const float*)d_in[12];
    float* out = (float*)d_out;              // PI[0..N-1], V at out[N]

    const int N = N_NODES, E = N_EDGES, F = HID_F;
    const size_t bufElems = (size_t)N * F;                    // 2.56M floats
    const size_t bufBytes = bufElems * sizeof(float);         // 10.24 MB

    // workspace layout (256-byte aligned slices)
    char* ws = (char*)d_ws;
    float* t    = (float*)(ws);                               // hW   [N,F]
    float* h    = (float*)(ws + bufBytes);                    // agg/h[N,F]
    float* deg  = (float*)(ws + 2 * bufBytes);                // [N]
    float* invd = (float*)(ws + 2 * bufBytes + 40192);        // [N]
    float* part = (float*)(ws + 2 * bufBytes + 2 * 40192);    // [64, F]

    const int P = 64;  // colsum partial blocks

    // --- degree / edge norm ---
    hipMemsetAsync(deg, 0, (size_t)N * sizeof(float), stream);
    degree_kernel<<<(E + 255) / 256, 256, 0, stream>>>(src, deg, E);
    invdeg_kernel<<<(N + 255) / 256, 256, 0, stream>>>(deg, invd, N);

    // wave-tasks = (M/16) * (F/64) = 625 * 4 = 2500 ; 8 waves per block
    const int gemmBlocks = ((N / 16) * (F / 64) + 7) / 8;     // 313
    const int biasBlocks = (int)((bufElems + 255) / 256);
    const int scatBlocks = (E + 3) / 4;

    // --- layer 1: t = features @ W1 ; h = scatter(t) ; relu(h + b1) ---
    gemm_wmma_f32<<<gemmBlocks, 256, 0, stream>>>(features, W1, t, N, IN_F, F);
    hipMemsetAsync(h, 0, bufBytes, stream);
    edge_scatter<<<scatBlocks, 256, 0, stream>>>(t, src, dst, invd, h, E, F);
    bias_act_kernel<<<biasBlocks, 256, 0, stream>>>(h, b1, (int)bufElems, F, 1);

    // --- layer 2 ---
    gemm_wmma_f32<<<gemmBlocks, 256, 0, stream>>>(h, W2, t, N, F, F);
    hipMemsetAsync(h, 0, bufBytes, stream);
    edge_scatter<<<scatBlocks, 256, 0, stream>>>(t, src, dst, invd, h, E, F);
    bias_act_kernel<<<biasBlocks, 256, 0, stream>>>(h, b2, (int)bufElems, F, 1);

    // --- layer 3 (no activation) ---
    gemm_wmma_f32<<<gemmBlocks, 256, 0, stream>>>(h, W3, t, N, F, F);
    hipMemsetAsync(h, 0, bufBytes, stream);
    edge_scatter<<<scatBlocks, 256, 0, stream>>>(t, src, dst, invd, h, E, F);
    bias_act_kernel<<<biasBlocks, 256, 0, stream>>>(h, b3, (int)bufElems, F, 0);

    // --- heads ---
    colsum_partial<<<P, 256, 0, stream>>>(h, part, N, F);
    value_head<<<1, 256, 0, stream>>>(part, P, Wv, bv, out + N, N, F);
    policy_head<<<(N + 7) / 8, 256, 0, stream>>>(h, Wp, bp, out, N, F);
}